// DSSGINConv_38293928411680
// MI455X (gfx1250) — compile-verified
//
#include <hip/hip_runtime.h>

typedef __attribute__((ext_vector_type(2))) float v2f;
typedef __attribute__((ext_vector_type(8))) float v8f;

#define NB 128    // subgraphs
#define NN 2048   // nodes
#define ND 64     // emb dim
#define NE 32768  // edges

// ---------------- CSR build ----------------
__global__ void k_zero_u32(unsigned* p, int n) {
  int i = blockIdx.x * blockDim.x + threadIdx.x;
  if (i < n) p[i] = 0u;
}

__global__ void k_count(const int* __restrict__ dst, unsigned* __restrict__ deg) {
  int e = blockIdx.x * blockDim.x + threadIdx.x;
  if (e < NE) atomicAdd(&deg[dst[e]], 1u);
}

// single-block exclusive scan over 2048 degrees -> off[0..2048], cursor[i]=off[i]
__global__ void k_scan(const unsigned* __restrict__ deg, unsigned* __restrict__ off,
                       unsigned* __restrict__ cursor) {
  __shared__ unsigned s0[NN];
  __shared__ unsigned s1[NN];
  int t = threadIdx.x;  // 256 threads
  for (int i = t; i < NN; i += 256) s0[i] = deg[i];
  __syncthreads();
  unsigned* a = s0;
  unsigned* b = s1;
  for (int st = 1; st < NN; st <<= 1) {
    for (int i = t; i < NN; i += 256)
      b[i] = a[i] + ((i >= st) ? a[i - st] : 0u);
    __syncthreads();
    unsigned* tmp = a; a = b; b = tmp;
  }
  // a now holds the inclusive scan
  for (int i = t; i < NN; i += 256) {
    off[i + 1] = a[i];
    cursor[i] = (i == 0) ? 0u : a[i - 1];
  }
  if (t == 0) off[0] = 0u;
}

__global__ void k_fill(const int* __restrict__ src, const int* __restrict__ dst,
                       unsigned* __restrict__ cursor, unsigned* __restrict__ csr) {
  int e = blockIdx.x * blockDim.x + threadIdx.x;
  if (e < NE) {
    unsigned p = atomicAdd(&cursor[dst[e]], 1u);
    csr[p] = (unsigned)src[e];
  }
}

// ---------------- max over subgraphs ----------------
__global__ void k_maxb(const float* __restrict__ X, float* __restrict__ xmax) {
  int i = blockIdx.x * blockDim.x + threadIdx.x;  // over N*D
  float m = -3.4028234663852886e38f;
  for (int b = 0; b < NB; ++b)
    m = fmaxf(m, X[(size_t)b * (NN * ND) + i]);
  xmax[i] = m;
}

// ---------------- fused 2-layer MLP (Linear->ReLU->Linear->ReLU) via WMMA ----
// One wave computes one 16(M)x16(N) output tile; 4 waves/block cover 16 rows x
// all 64 columns. K=64 done as 16 chained V_WMMA_F32_16X16X4_F32 per layer.
// A 16x4 frag: lane l holds rows m=l&15, k = kbase + 2*(l>>4) + {0,1} (2 VGPRs)
// B 4x16 frag: lane l holds cols n=l&15, k = kbase + 2*(l>>4) + {0,1} (2 VGPRs)
// C/D 16x16:   VGPR j, lane l -> row j + 8*(l>>4), col l&15
__global__ void __launch_bounds__(128)
k_mlp2_wmma(const float* __restrict__ Xin, const float* __restrict__ W1,
            const float* __restrict__ b1, const float* __restrict__ W2,
            const float* __restrict__ b2, float* __restrict__ Yout) {
  __shared__ float Ys[16 * ND];  // layer-1 activations for this 16-row stripe
  const int lane = threadIdx.x & 31;
  const int wave = threadIdx.x >> 5;  // 0..3 -> output column tile
  const int col0 = wave * 16;
  const int hm = lane & 15;           // A-row / B-col index
  const int hk = lane >> 4;           // k-pair selector
  const size_t row0 = (size_t)blockIdx.x * 16;

  // ---- layer 1: preload W1 B-fragments (register-resident, L2-hot loads)
  v2f bw[16];
#pragma unroll
  for (int kk = 0; kk < 16; ++kk) {
    int k = kk * 4 + 2 * hk;
    bw[kk].x = W1[(k + 0) * ND + col0 + hm];
    bw[kk].y = W1[(k + 1) * ND + col0 + hm];
  }

  v8f acc;
  {
    float bv = b1[col0 + hm];  // bias is per-column: same for all 8 C regs
#pragma unroll
    for (int j = 0; j < 8; ++j) acc[j] = bv;
  }

  const float* xrow = Xin + (row0 + hm) * ND;
#pragma unroll
  for (int kk = 0; kk < 16; ++kk) {
    int k = kk * 4 + 2 * hk;
    v2f a;
    a.x = xrow[k];
    a.y = xrow[k + 1];
    acc = __builtin_amdgcn_wmma_f32_16x16x4_f32(false, a, false, bw[kk],
                                                (short)0, acc, false, false);
  }
  // ReLU + stage Y tile to LDS (row-major 16x64)
#pragma unroll
  for (int j = 0; j < 8; ++j) {
    Ys[(j + 8 * hk) * ND + col0 + hm] = fmaxf(acc[j], 0.0f);
  }
  __syncthreads();

  // ---- layer 2: A-fragments come from LDS, B from W2
#pragma unroll
  for (int kk = 0; kk < 16; ++kk) {
    int k = kk * 4 + 2 * hk;
    bw[kk].x = W2[(k + 0) * ND + col0 + hm];
    bw[kk].y = W2[(k + 1) * ND + col0 + hm];
  }
  {
    float bv = b2[col0 + hm];
#pragma unroll
    for (int j = 0; j < 8; ++j) acc[j] = bv;
  }
#pragma unroll
  for (int kk = 0; kk < 16; ++kk) {
    int k = kk * 4 + 2 * hk;
    v2f a;
    a.x = Ys[hm * ND + k];
    a.y = Ys[hm * ND + k + 1];
    acc = __builtin_amdgcn_wmma_f32_16x16x4_f32(false, a, false, bw[kk],
                                                (short)0, acc, false, false);
  }
#pragma unroll
  for (int j = 0; j < 8; ++j) {
    Yout[(row0 + j + 8 * hk) * ND + col0 + hm] = fmaxf(acc[j], 0.0f);
  }
}

// ---------------- CSR gathers (no fp atomics) ----------------
// agg[n,d] = sum over incoming edges of nodex[src,d]
__global__ void k_gather_node(const float* __restrict__ nodex,
                              const unsigned* __restrict__ off,
                              const unsigned* __restrict__ csr,
                              float* __restrict__ agg) {
  int r = blockIdx.x * 4 + (threadIdx.x >> 6);  // node id
  int d = threadIdx.x & 63;
  unsigned s = off[r], e = off[r + 1];
  float acc = 0.0f;
  for (unsigned j = s; j < e; ++j) acc += nodex[(size_t)csr[j] * ND + d];
  agg[(size_t)r * ND + d] = acc;
}

// out[b,n,d] = agg[n,d] + sum over incoming edges of h[b,src,d]
// 4 b's per block share one n -> CSR range is block-uniform (scalarizes).
__global__ void k_gather_final(const float* __restrict__ h,
                               const float* __restrict__ agg,
                               const unsigned* __restrict__ off,
                               const unsigned* __restrict__ csr,
                               float* __restrict__ out) {
  int n = blockIdx.x & (NN - 1);
  int b = ((blockIdx.x >> 11) << 2) + (threadIdx.x >> 6);
  int d = threadIdx.x & 63;
  unsigned s = off[n], e = off[n + 1];
  float acc = agg[(size_t)n * ND + d];
  const float* hb = h + (size_t)b * NN * ND;
  for (unsigned j = s; j < e; ++j) acc += hb[(size_t)csr[j] * ND + d];
  out[((size_t)b * NN + n) * ND + d] = acc;
}

extern "C" void kernel_launch(void* const* d_in, const int* in_sizes, int n_in,
                              void* d_out, int out_size, void* d_ws, size_t ws_size,
                              hipStream_t stream) {
  (void)in_sizes; (void)n_in; (void)out_size; (void)ws_size;
  const float* X   = (const float*)d_in[0];
  const int*   ei  = (const int*)d_in[1];
  const float* W1t = (const float*)d_in[2];
  const float* b1t = (const float*)d_in[3];
  const float* W2t = (const float*)d_in[4];
  const float* b2t = (const float*)d_in[5];
  const float* W1n = (const float*)d_in[6];
  const float* b1n = (const float*)d_in[7];
  const float* W2n = (const float*)d_in[8];
  const float* b2n = (const float*)d_in[9];
  float* out = (float*)d_out;

  // workspace layout (all 4-byte types): ~65.6 MB
  float* h     = (float*)d_ws;                    // [B*N*D] = 64 MB (L2-resident)
  float* xmax  = h + (size_t)NB * NN * ND;        // [N*D]
  float* nodex = xmax + (size_t)NN * ND;          // [N*D]
  float* agg   = nodex + (size_t)NN * ND;         // [N*D]
  unsigned* deg    = (unsigned*)(agg + (size_t)NN * ND);  // [N]
  unsigned* off    = deg + NN;                            // [N+1]
  unsigned* cursor = off + (NN + 1);                      // [N]
  unsigned* csr    = cursor + NN;                         // [E]

  const int* esrc = ei;       // edge_index[0]
  const int* edst = ei + NE;  // edge_index[1]

  // CSR build (int atomics only)
  k_zero_u32<<<(NN + 255) / 256, 256, 0, stream>>>(deg, NN);
  k_count<<<NE / 256, 256, 0, stream>>>(edst, deg);
  k_scan<<<1, 256, 0, stream>>>(deg, off, cursor);
  k_fill<<<NE / 256, 256, 0, stream>>>(esrc, edst, cursor, csr);

  // DSS node branch: max over B -> MLP -> edge aggregation
  k_maxb<<<(NN * ND) / 256, 256, 0, stream>>>(X, xmax);
  k_mlp2_wmma<<<NN / 16, 128, 0, stream>>>(xmax, W1n, b1n, W2n, b2n, nodex);
  k_gather_node<<<NN / 4, 256, 0, stream>>>(nodex, off, csr, agg);

  // NestedConv branch: big fused MLP, then per-(b,dst) gather fused with broadcast add
  k_mlp2_wmma<<<(NB * NN) / 16, 128, 0, stream>>>(X, W1t, b1t, W2t, b2t, h);
  k_gather_final<<<(NB / 4) * NN, 256, 0, stream>>>(h, agg, off, csr, out);
}